// GMWA_80522046865464
// MI455X (gfx1250) — compile-verified
//
#include <hip/hip_runtime.h>
#include <hip/hip_bf16.h>

// ---------------------------------------------------------------------------
// GATv2 x2 + patch mapper, CDNA5 (gfx1250), wave32.
// WMMA f32 16x16x4 for all GEMMs; TDM (tensor_load_to_lds) double-buffered
// tile staging for the big GEMM; VALU for the GATv2 nonlinearity/softmax.
// ---------------------------------------------------------------------------

typedef float v2f __attribute__((ext_vector_type(2)));
typedef float v8f __attribute__((ext_vector_type(8)));
typedef unsigned int v4u __attribute__((ext_vector_type(4)));
typedef int v4i __attribute__((ext_vector_type(4)));
typedef int v8i __attribute__((ext_vector_type(8)));

#define NN      768
#define WAVK    6360
#define HEADS   4
#define HID     32
#define GDIM    128          // HEADS*HID
#define PP      8
#define SUP     96
#define NEG_SLOPE 0.2f

#define TM 64
#define TN 32
#define TK 32

#if defined(__AMDGCN__) && __has_builtin(__builtin_amdgcn_tensor_load_to_lds) && \
    __has_builtin(__builtin_amdgcn_s_wait_tensorcnt)
#define USE_TDM 1
#else
#define USE_TDM 0
#endif

#if USE_TDM
typedef __attribute__((address_space(3))) void lds_void;
#define LDS_OFF(p) ((unsigned)(unsigned long long)(lds_void*)(p))

// Issue a 2D TDM tile load: global (row-major, stride0 elems/row) -> LDS.
// data_size = 4B. OOB reads (beyond tensor_d0/d1 from the tile origin) are
// zero-filled by hardware (== our K-tail padding). pad: +1 DWORD after every
// 32 DWORDs stored (pad_interval code 4, pad_amount code 0) -> LDS row
// stride 33 floats, killing bank conflicts on the A-fragment reads.
__device__ __forceinline__ void tdm_load_2d(unsigned lds_addr, const void* gptr,
                                            unsigned tile_d0, unsigned tile_d1,
                                            unsigned tens_d0, unsigned tens_d1,
                                            unsigned stride0)
{
    unsigned long long ga = (unsigned long long)gptr;
    v4u g0;
    g0[0] = 1u;                                   // count=1, user mode
    g0[1] = lds_addr;                             // LDS byte address
    g0[2] = (unsigned)(ga & 0xFFFFFFFFull);       // global_addr[31:0]
    g0[3] = (unsigned)((ga >> 32) & 0x01FFFFFFull) | (2u << 30); // [56:32] | type=2
    v8i g1;
    g1[0] = (int)((2u << 16)        // data_size = 4B
                | (1u << 20)        // pad_enable
                | (4u << 22)        // pad_interval: 32 DWORDs
                | (0u << 25));      // pad_amount: 1 DWORD
    g1[1] = (int)((tens_d0 & 0xFFFFu) << 16);                       // dim0[15:0]
    g1[2] = (int)(((tens_d0 >> 16) & 0xFFFFu) | ((tens_d1 & 0xFFFFu) << 16));
    g1[3] = (int)(((tens_d1 >> 16) & 0xFFFFu) | (tile_d0 << 16));
    g1[4] = (int)(tile_d1 & 0xFFFFu);                               // tile_dim1 (dim2=0)
    g1[5] = (int)stride0;                                           // dim0_stride lo
    g1[6] = 0;                                                      // stride hi / dim1_stride
    g1[7] = 0;
    v4i z4 = {0, 0, 0, 0};
#if __clang_major__ >= 23
    v8i z8 = {0, 0, 0, 0, 0, 0, 0, 0};
    __builtin_amdgcn_tensor_load_to_lds(g0, g1, z4, z4, z8, 0);
#else
    __builtin_amdgcn_tensor_load_to_lds(g0, g1, z4, z4, 0);
#endif
}
#endif // USE_TDM

// ---------------- WMMA GEMM: C[M,N] = A[M,K] @ B[K,N], all row-major -------
#if USE_TDM
__global__ __launch_bounds__(256)
void gemm_wmma_f32(const float* __restrict__ A, const float* __restrict__ B,
                   float* __restrict__ C, int M, int N, int K)
{
    // TDM writes rows at (TK+1)-float stride via hardware padding.
    __shared__ float As[2][TM][TK + 1];
    __shared__ float Bs[2][TK][TN + 1];

    const int tid  = threadIdx.x;
    const int lane = tid & 31;
    const int wave = tid >> 5;          // 8 waves
    const int wm   = wave >> 1;         // 0..3  -> 16-row slice
    const int wn   = wave & 1;          // 0..1  -> 16-col slice
    const int bm   = blockIdx.y * TM;
    const int bn   = blockIdx.x * TN;
    const int l16  = lane & 15;
    const int lhi  = lane >> 4;

    const unsigned ldsA[2] = { LDS_OFF(&As[0][0][0]), LDS_OFF(&As[1][0][0]) };
    const unsigned ldsB[2] = { LDS_OFF(&Bs[0][0][0]), LDS_OFF(&Bs[1][0][0]) };

    const int ntiles = (K + TK - 1) / TK;

    if (wave == 0) {   // one wave drives the DMA engine (EXEC ignored by TDM)
        tdm_load_2d(ldsA[0], A + (size_t)bm * K, TK, TM,
                    (unsigned)K, (unsigned)(M - bm), (unsigned)K);
        tdm_load_2d(ldsB[0], B + bn, TN, TK,
                    (unsigned)(N - bn), (unsigned)K, (unsigned)N);
    }

    v8f acc = {};
    for (int t = 0; t < ntiles; ++t) {
        const int buf = t & 1;
        if (wave == 0) {
            if (t + 1 < ntiles) {
                const int k1 = (t + 1) * TK;
                tdm_load_2d(ldsA[buf ^ 1], A + (size_t)bm * K + k1, TK, TM,
                            (unsigned)(K - k1), (unsigned)(M - bm), (unsigned)K);
                tdm_load_2d(ldsB[buf ^ 1], B + (size_t)k1 * N + bn, TN, TK,
                            (unsigned)(N - bn), (unsigned)(K - k1), (unsigned)N);
                // 2 newly-issued ops may remain in flight; current tile done.
                __builtin_amdgcn_s_wait_tensorcnt((short)2);
            } else {
                __builtin_amdgcn_s_wait_tensorcnt((short)0);
            }
        }
        __syncthreads();

        const int am   = wm * 16 + l16;
        const int bcol = wn * 16 + l16;
        #pragma unroll
        for (int kk = 0; kk < TK; kk += 4) {
            v2f a, b;
            // A 16x4 frag: lanes0-15 K={kk,kk+1}, lanes16-31 K={kk+2,kk+3}
            a[0] = As[buf][am][kk + 2 * lhi];
            a[1] = As[buf][am][kk + 2 * lhi + 1];
            // B 4x16 frag: VGPR0: K=kk(lo)/kk+2(hi); VGPR1: +1
            b[0] = Bs[buf][kk + 2 * lhi][bcol];
            b[1] = Bs[buf][kk + 2 * lhi + 1][bcol];
            acc = __builtin_amdgcn_wmma_f32_16x16x4_f32(
                false, a, false, b, (short)0, acc, false, false);
        }
        __syncthreads();
    }

    const int ccol = bn + wn * 16 + l16;
    #pragma unroll
    for (int v = 0; v < 8; ++v) {
        int crow = bm + wm * 16 + v + 8 * lhi;
        C[crow * N + ccol] = acc[v];
    }
}
#else  // !USE_TDM : cooperative-load fallback (compiles everywhere)
__global__ __launch_bounds__(256)
void gemm_wmma_f32(const float* __restrict__ A, const float* __restrict__ B,
                   float* __restrict__ C, int M, int N, int K)
{
    __shared__ float As[TM][TK + 1];
    __shared__ float Bs[TK][TN + 1];

    const int tid  = threadIdx.x;
    const int lane = tid & 31;
    const int wave = tid >> 5;
    const int wm   = wave >> 1;
    const int wn   = wave & 1;
    const int bm   = blockIdx.y * TM;
    const int bn   = blockIdx.x * TN;
    const int l16  = lane & 15;
    const int lhi  = lane >> 4;

    v8f acc = {};

    for (int k0 = 0; k0 < K; k0 += TK) {
        for (int idx = tid; idx < TM * TK; idx += 256) {
            int m = idx / TK, k = idx % TK;
            int gk = k0 + k;
            As[m][k] = (gk < K) ? A[(bm + m) * K + gk] : 0.0f;
        }
        for (int idx = tid; idx < TK * TN; idx += 256) {
            int k = idx / TN, n = idx % TN;
            int gk = k0 + k;
            Bs[k][n] = (gk < K) ? B[gk * N + (bn + n)] : 0.0f;
        }
        __syncthreads();

        const int am = wm * 16 + l16;
        const int bcol = wn * 16 + l16;
        #pragma unroll
        for (int kk = 0; kk < TK; kk += 4) {
            v2f a, b;
            a[0] = As[am][kk + 2 * lhi];
            a[1] = As[am][kk + 2 * lhi + 1];
            b[0] = Bs[kk + 2 * lhi][bcol];
            b[1] = Bs[kk + 2 * lhi + 1][bcol];
            acc = __builtin_amdgcn_wmma_f32_16x16x4_f32(
                false, a, false, b, (short)0, acc, false, false);
        }
        __syncthreads();
    }

    const int ccol = bn + wn * 16 + l16;
    #pragma unroll
    for (int v = 0; v < 8; ++v) {
        int crow = bm + wm * 16 + v + 8 * lhi;
        C[crow * N + ccol] = acc[v];
    }
}
#endif // USE_TDM

// ---------------- GATv2 attention: masked softmax over j -------------------
__global__ __launch_bounds__(256)
void gatv2_attn(const float* __restrict__ g_l, const float* __restrict__ g_r,
                const float* __restrict__ avec, const int* __restrict__ adj,
                float* __restrict__ att, int n)
{
    const int i   = blockIdx.x;
    const int tid = threadIdx.x;
    __shared__ float s_gr[GDIM];
    __shared__ float s_a[HID];
    __shared__ float s_red[HEADS][256];

    if (tid < GDIM) s_gr[tid] = g_r[i * GDIM + tid];
    if (tid < HID)  s_a[tid]  = avec[tid];
    __syncthreads();

    float e_loc[4][HEADS];     // up to 4 j's per thread (n<=1024)
    float lmax[HEADS];
    #pragma unroll
    for (int h = 0; h < HEADS; ++h) lmax[h] = -1e30f;

    int nIter = 0;
    for (int j = tid; j < n; j += 256, ++nIter) {
        const int m = adj[i * n + j];
        if (m != 0) {
            const float* gl = g_l + j * GDIM;
            #pragma unroll
            for (int h = 0; h < HEADS; ++h) {
                float acc = 0.0f;
                #pragma unroll
                for (int f = 0; f < HID; ++f) {
                    float x  = s_gr[h * HID + f] + gl[h * HID + f];
                    float lr = (x > 0.0f) ? x : NEG_SLOPE * x;
                    acc = fmaf(lr, s_a[f], acc);
                }
                e_loc[nIter][h] = acc;
                lmax[h] = fmaxf(lmax[h], acc);
            }
        } else {
            #pragma unroll
            for (int h = 0; h < HEADS; ++h) e_loc[nIter][h] = -1e30f;
        }
    }

    #pragma unroll
    for (int h = 0; h < HEADS; ++h) s_red[h][tid] = lmax[h];
    __syncthreads();
    for (int s = 128; s > 0; s >>= 1) {
        if (tid < s) {
            #pragma unroll
            for (int h = 0; h < HEADS; ++h)
                s_red[h][tid] = fmaxf(s_red[h][tid], s_red[h][tid + s]);
        }
        __syncthreads();
    }
    float mh[HEADS];
    #pragma unroll
    for (int h = 0; h < HEADS; ++h) mh[h] = s_red[h][0];
    __syncthreads();

    float lsum[HEADS];
    #pragma unroll
    for (int h = 0; h < HEADS; ++h) lsum[h] = 0.0f;
    {
        int it = 0;
        for (int j = tid; j < n; j += 256, ++it) {
            #pragma unroll
            for (int h = 0; h < HEADS; ++h) {
                float p = __expf(e_loc[it][h] - mh[h]);   // masked -> 0
                e_loc[it][h] = p;
                lsum[h] += p;
            }
        }
    }
    #pragma unroll
    for (int h = 0; h < HEADS; ++h) s_red[h][tid] = lsum[h];
    __syncthreads();
    for (int s = 128; s > 0; s >>= 1) {
        if (tid < s) {
            #pragma unroll
            for (int h = 0; h < HEADS; ++h)
                s_red[h][tid] += s_red[h][tid + s];
        }
        __syncthreads();
    }
    float rinv[HEADS];
    #pragma unroll
    for (int h = 0; h < HEADS; ++h) rinv[h] = 1.0f / s_red[h][0];

    {
        int it = 0;
        for (int j = tid; j < n; j += 256, ++it) {
            #pragma unroll
            for (int h = 0; h < HEADS; ++h)
                att[(i * n + j) * HEADS + h] = e_loc[it][h] * rinv[h];
        }
    }
}

// ---------------- aggregation: res[i,h,f] = sum_j att[i,j,h]*g_r[j,h,f] ----
__global__ __launch_bounds__(256)
void gat_agg_wmma(const float* __restrict__ att, const float* __restrict__ g_r,
                  float* __restrict__ res, int n)
{
    const int tid  = threadIdx.x;
    const int lane = tid & 31;
    const int wave = tid >> 5;
    const int h    = wave >> 1;            // 0..3
    const int f0   = (wave & 1) * 16;      // 0 or 16
    const int i0   = blockIdx.x * 16;
    const int l16  = lane & 15;
    const int lhi  = lane >> 4;

    v8f acc = {};
    const int row = i0 + l16;
    const int fc  = f0 + l16;
    for (int j = 0; j < n; j += 4) {
        v2f a, b;
        const int ka = j + 2 * lhi;
        a[0] = att[(row * n + ka) * HEADS + h];
        a[1] = att[(row * n + ka + 1) * HEADS + h];
        b[0] = g_r[ka * GDIM + h * HID + fc];
        b[1] = g_r[(ka + 1) * GDIM + h * HID + fc];
        acc = __builtin_amdgcn_wmma_f32_16x16x4_f32(
            false, a, false, b, (short)0, acc, false, false);
    }
    #pragma unroll
    for (int v = 0; v < 8; ++v) {
        int i = i0 + v + 8 * lhi;
        res[(i * HEADS + h) * HID + f0 + l16] = acc[v];
    }
}

// ---------------- mean over heads (+ optional ELU) -------------------------
__global__ __launch_bounds__(256)
void mean_elu(const float* __restrict__ res, float* __restrict__ z,
              int n, int do_elu)
{
    int idx = blockIdx.x * 256 + threadIdx.x;
    if (idx >= n * HID) return;
    int i = idx >> 5, f = idx & 31;
    float s = 0.25f * (res[(i * HEADS + 0) * HID + f] +
                       res[(i * HEADS + 1) * HID + f] +
                       res[(i * HEADS + 2) * HID + f] +
                       res[(i * HEADS + 3) * HID + f]);
    if (do_elu) s = (s > 0.0f) ? s : (__expf(s) - 1.0f);
    z[idx] = s;
}

// ---------------- patch mapper ---------------------------------------------
__global__ __launch_bounds__(256)
void mapper(const float* __restrict__ z, const float* __restrict__ Wm,
            const float* __restrict__ bm, float* __restrict__ sta,
            float* __restrict__ z_hemo)
{
    int idx = blockIdx.x * 256 + threadIdx.x;   // over P*32*96 = 24576
    if (idx >= PP * HID * SUP) return;
    int p = idx / (HID * SUP);
    int r = idx % (HID * SUP);
    int t = r / SUP;
    int e = r % SUP;

    float acc = bm[p * SUP + e];
    const float* wmp = Wm + p * SUP * SUP;
    for (int d = 0; d < SUP; ++d)
        acc = fmaf(z[(p * SUP + d) * HID + t], wmp[d * SUP + e], acc);

    sta[t * (PP * SUP) + p * SUP + e] = acc;      // [32, 8, 96]
    z_hemo[(p * SUP + e) * HID + t]   = acc;      // [768, 32]
}

// ---------------------------------------------------------------------------
extern "C" void kernel_launch(void* const* d_in, const int* in_sizes, int n_in,
                              void* d_out, int out_size, void* d_ws, size_t ws_size,
                              hipStream_t stream)
{
    const float* x_ele = (const float*)d_in[0];
    const float* Wl1   = (const float*)d_in[1];
    const float* Wr1   = (const float*)d_in[2];
    const float* a1    = (const float*)d_in[3];
    const float* Wm    = (const float*)d_in[4];
    const float* bm    = (const float*)d_in[5];
    const float* Wl2   = (const float*)d_in[6];
    const float* Wr2   = (const float*)d_in[7];
    const float* a2    = (const float*)d_in[8];
    const int*   adjS  = (const int*)d_in[9];
    const int*   adjT  = (const int*)d_in[10];

    float* out = (float*)d_out;
    float* ws  = (float*)d_ws;

    float* g_l1   = ws;                    // 768*128
    float* g_r1   = ws + 98304;            // 768*128
    float* res    = ws + 196608;           // 768*4*32 (reused)
    float* z      = ws + 294912;           // 768*32
    float* z_hemo = ws + 319488;           // 768*32
    float* g_l2   = ws + 344064;           // 768*128
    float* g_r2   = ws + 442368;           // 768*128

    float* o_out  = out;                       // 768*32
    float* o_satt = out + 24576;               // 768*768*4
    float* o_tatt = out + 24576 + 2359296;     // 768*768*4
    float* o_sta  = out + 24576 + 2 * 2359296; // 32*8*96

    dim3 gGemm(GDIM / TN, NN / TM);            // (4, 12)

    gemm_wmma_f32<<<gGemm, 256, 0, stream>>>(x_ele, Wl1, g_l1, NN, GDIM, WAVK);
    gemm_wmma_f32<<<gGemm, 256, 0, stream>>>(x_ele, Wr1, g_r1, NN, GDIM, WAVK);
    gatv2_attn<<<NN, 256, 0, stream>>>(g_l1, g_r1, a1, adjS, o_satt, NN);
    gat_agg_wmma<<<NN / 16, 256, 0, stream>>>(o_satt, g_r1, res, NN);
    mean_elu<<<(NN * HID + 255) / 256, 256, 0, stream>>>(res, z, NN, 1);

    mapper<<<(PP * HID * SUP + 255) / 256, 256, 0, stream>>>(z, Wm, bm, o_sta, z_hemo);

    gemm_wmma_f32<<<gGemm, 256, 0, stream>>>(z_hemo, Wl2, g_l2, NN, GDIM, HID);
    gemm_wmma_f32<<<gGemm, 256, 0, stream>>>(z_hemo, Wr2, g_r2, NN, GDIM, HID);
    gatv2_attn<<<NN, 256, 0, stream>>>(g_l2, g_r2, a2, adjT, o_tatt, NN);
    gat_agg_wmma<<<NN / 16, 256, 0, stream>>>(o_tatt, g_r2, res, NN);
    mean_elu<<<(NN * HID + 255) / 256, 256, 0, stream>>>(res, o_out, NN, 0);
}